// DFAGNN_minus_max_68375879353082
// MI455X (gfx1250) — compile-verified
//
#include <hip/hip_runtime.h>
#include <math.h>

// Problem constants (match reference): B=4, N=10000, E=80000, IP=4, H=64, EF=64
static constexpr int Bc  = 4;
static constexpr int Nc  = 10000;
static constexpr int Ec  = 80000;
static constexpr int IPc = 4;
static constexpr int Hc  = 64;

static constexpr int ROWS = Bc * Nc * IPc;          // 160000 rows of H=64
static constexpr size_t NH_ELEMS = (size_t)ROWS * Hc; // 10,240,000 floats

typedef float v2f __attribute__((ext_vector_type(2)));
typedef float v8f __attribute__((ext_vector_type(8)));

#define WMMA_F32(a, b, c) \
    __builtin_amdgcn_wmma_f32_16x16x4_f32(false, (a), false, (b), (short)0, (c), false, false)

// ---------------------------------------------------------------------------
// K1: nh_fused = [node_fts | hidden] @ W_nh + b_nh     (160000x128 @ 128x64)
// 128 threads = 4 waves; each wave computes a 16x64 output tile via
// v_wmma_f32_16x16x4_f32 (4 column tiles x 32 K-steps).
// A rows staged in LDS with stride 132 -> conflict-free ds_load_b64.
// ---------------------------------------------------------------------------
__global__ __launch_bounds__(128) void k_nh_fused(
    const float* __restrict__ node_fts, const float* __restrict__ hidden,
    const float* __restrict__ W_nh, const float* __restrict__ b_nh,
    float* __restrict__ nh_out)
{
    __shared__ float sA[4][16 * 132];   // per-wave 16 rows x 128 cols (pad 132)

    const int tid  = threadIdx.x;
    const int wave = tid >> 5;
    const int lane = tid & 31;
    const long long rowbase = (long long)blockIdx.x * 64 + wave * 16;

    // Stage A: lanes 0-15 load node_fts cols, lanes 16-31 load hidden cols.
    {
        const float* srcp = (lane < 16) ? node_fts : hidden;
        const int c4 = (lane & 15) * 4;
        const int halfofs = (lane >> 4) * 64;
        for (int r = 0; r < 16; ++r) {
            const float4 v = *(const float4*)(srcp + (size_t)(rowbase + r) * Hc + c4);
            float* dst = &sA[wave][r * 132 + halfofs + c4];
            dst[0] = v.x; dst[1] = v.y; dst[2] = v.z; dst[3] = v.w;
        }
    }
    __syncthreads();

    const int m   = lane & 15;          // A row within tile
    const int klo = (lane >> 4) * 2;    // K sub-offset (0 or 2)
    const int nn  = lane & 15;          // B/D column within tile

    v8f acc0 = {}, acc1 = {}, acc2 = {}, acc3 = {};
    for (int k0 = 0; k0 < 128; k0 += 4) {
        const float2 av = *(const float2*)(&sA[wave][m * 132 + k0 + klo]);
        v2f a; a.x = av.x; a.y = av.y;

        const float* w0 = W_nh + (size_t)(k0 + klo) * Hc;      // row k
        const float* w1 = w0 + Hc;                              // row k+1
        v2f b0, b1, b2, b3;
        b0.x = w0[ 0 + nn]; b0.y = w1[ 0 + nn];
        b1.x = w0[16 + nn]; b1.y = w1[16 + nn];
        b2.x = w0[32 + nn]; b2.y = w1[32 + nn];
        b3.x = w0[48 + nn]; b3.y = w1[48 + nn];

        acc0 = WMMA_F32(a, b0, acc0);
        acc1 = WMMA_F32(a, b1, acc1);
        acc2 = WMMA_F32(a, b2, acc2);
        acc3 = WMMA_F32(a, b3, acc3);
    }

    // D layout: VGPR v -> row v (lanes 0-15) / row v+8 (lanes 16-31), col = lane&15
    const int rofs = (lane >> 4) * 8;
    for (int v = 0; v < 8; ++v) {
        float* orow = nh_out + (size_t)(rowbase + v + rofs) * Hc;
        orow[ 0 + nn] = acc0[v] + b_nh[ 0 + nn];
        orow[16 + nn] = acc1[v] + b_nh[16 + nn];
        orow[32 + nn] = acc2[v] + b_nh[32 + nn];
        orow[48 + nn] = acc3[v] + b_nh[48 + nn];
    }
}

// ---------------------------------------------------------------------------
// K2: fill aggregation buffer with -inf (segment_max identity)
// ---------------------------------------------------------------------------
__global__ __launch_bounds__(256) void k_init_neg_inf(float4* __restrict__ p)
{
    const float ninf = -__builtin_inff();
    p[(size_t)blockIdx.x * 256 + threadIdx.x] = make_float4(ninf, ninf, ninf, ninf);
}

// ---------------------------------------------------------------------------
// K3: per-edge coeff + gather + scatter-max.  One wave32 per edge.
// coeff = dot(edge_fts[e], W_e) + b_e  (wave reduction)
// then 256 floats: load nh_fused[src], scale, atomic-max into agg[tgt].
// Float max via signed-max (v>=0) / unsigned-min (v<0) bit trick: one
// hardware atomic per element, order-independent -> deterministic.
// ---------------------------------------------------------------------------
__global__ __launch_bounds__(256) void k_edge_scatter(
    const int* __restrict__ cfg, const float* __restrict__ edge_fts,
    const float* __restrict__ W_e, const float* __restrict__ b_e,
    const float* __restrict__ nh, float* __restrict__ agg)
{
    const int lane = threadIdx.x & 31;
    const int eg   = blockIdx.x * 8 + (threadIdx.x >> 5);   // 0 .. B*E-1
    const int b    = eg / Ec;

    float partial = edge_fts[(size_t)eg * 64 + lane]      * W_e[lane]
                  + edge_fts[(size_t)eg * 64 + 32 + lane] * W_e[32 + lane];
    #pragma unroll
    for (int off = 16; off > 0; off >>= 1)
        partial += __shfl_xor(partial, off, 32);
    const float coeff = partial + b_e[0];

    const int src = cfg[(size_t)eg * 2 + 0];
    const int tgt = cfg[(size_t)eg * 2 + 1];
    const float* srow = nh  + ((size_t)b * Nc + src) * (IPc * Hc);
    float*       trow = agg + ((size_t)b * Nc + tgt) * (IPc * Hc);

    #pragma unroll
    for (int i = 0; i < 8; ++i) {
        const int j = lane + 32 * i;
        const float v = coeff * srow[j];
        if (v >= 0.0f) atomicMax((int*)(trow + j), __float_as_int(v));
        else           atomicMin((unsigned int*)(trow + j), __float_as_uint(v));
    }
}

// ---------------------------------------------------------------------------
// K4: Q/K/V = agg @ W_{q,k,v} + b  via WMMA, staged into padded LDS (stride 65),
// then per-node softmax attention over IP=4 and coalesced store.
// Block = 128 threads = 4 waves = 64 rows = 16 nodes.
// ---------------------------------------------------------------------------
__global__ __launch_bounds__(128) void k_qkv_attn(
    const float* __restrict__ agg,
    const float* __restrict__ Wq, const float* __restrict__ bq,
    const float* __restrict__ Wk, const float* __restrict__ bk,
    const float* __restrict__ Wv, const float* __restrict__ bv,
    float* __restrict__ out)
{
    __shared__ float sQ[64 * 65];
    __shared__ float sK[64 * 65];
    __shared__ float sV[64 * 65];
    __shared__ float sP[64][4];

    const int tid  = threadIdx.x;
    const int wave = tid >> 5;
    const int lane = tid & 31;
    const long long rowbase = (long long)blockIdx.x * 64 + wave * 16;

    const int m   = lane & 15;
    const int klo = (lane >> 4) * 2;
    const int nn  = lane & 15;
    const int rofs = (lane >> 4) * 8;

    for (int mat = 0; mat < 3; ++mat) {
        const float* Wm = (mat == 0) ? Wq : ((mat == 1) ? Wk : Wv);
        const float* bm = (mat == 0) ? bq : ((mat == 1) ? bk : bv);
        float*      dst = (mat == 0) ? sQ : ((mat == 1) ? sK : sV);

        v8f acc0 = {}, acc1 = {}, acc2 = {}, acc3 = {};
        for (int k0 = 0; k0 < 64; k0 += 4) {
            const float2 av = *(const float2*)(agg + (size_t)(rowbase + m) * Hc + k0 + klo);
            v2f a; a.x = av.x; a.y = av.y;

            const float* w0 = Wm + (size_t)(k0 + klo) * Hc;
            const float* w1 = w0 + Hc;
            v2f b0, b1, b2, b3;
            b0.x = w0[ 0 + nn]; b0.y = w1[ 0 + nn];
            b1.x = w0[16 + nn]; b1.y = w1[16 + nn];
            b2.x = w0[32 + nn]; b2.y = w1[32 + nn];
            b3.x = w0[48 + nn]; b3.y = w1[48 + nn];

            acc0 = WMMA_F32(a, b0, acc0);
            acc1 = WMMA_F32(a, b1, acc1);
            acc2 = WMMA_F32(a, b2, acc2);
            acc3 = WMMA_F32(a, b3, acc3);
        }
        for (int v = 0; v < 8; ++v) {
            const int row = wave * 16 + v + rofs;       // 0..63 (block-local)
            dst[row * 65 +  0 + nn] = acc0[v] + bm[ 0 + nn];
            dst[row * 65 + 16 + nn] = acc1[v] + bm[16 + nn];
            dst[row * 65 + 32 + nn] = acc2[v] + bm[32 + nn];
            dst[row * 65 + 48 + nn] = acc3[v] + bm[48 + nn];
        }
    }
    __syncthreads();

    // softmax over IP=4 per node (64 rows -> threads 0..63)
    if (tid < 64) {
        const int r = tid, node = r >> 2;
        const float* q  = &sQ[r * 65];
        const float* k0 = &sK[(node * 4 + 0) * 65];
        const float* k1 = &sK[(node * 4 + 1) * 65];
        const float* k2 = &sK[(node * 4 + 2) * 65];
        const float* k3 = &sK[(node * 4 + 3) * 65];
        float s0 = 0.f, s1 = 0.f, s2 = 0.f, s3 = 0.f;
        for (int h = 0; h < 64; ++h) {
            const float qh = q[h];
            s0 += qh * k0[h]; s1 += qh * k1[h];
            s2 += qh * k2[h]; s3 += qh * k3[h];
        }
        s0 *= 0.125f; s1 *= 0.125f; s2 *= 0.125f; s3 *= 0.125f;  // 1/sqrt(64)
        const float mx = fmaxf(fmaxf(s0, s1), fmaxf(s2, s3));
        const float p0 = __expf(s0 - mx), p1 = __expf(s1 - mx);
        const float p2 = __expf(s2 - mx), p3 = __expf(s3 - mx);
        const float inv = 1.0f / (p0 + p1 + p2 + p3);
        sP[r][0] = p0 * inv; sP[r][1] = p1 * inv;
        sP[r][2] = p2 * inv; sP[r][3] = p3 * inv;
    }
    __syncthreads();

    // out = P @ V : thread t covers row r=t>>1, 32 cols
    {
        const int r = tid >> 1, half = tid & 1, node = r >> 2;
        const int c0 = half * 32;
        const float p0 = sP[r][0], p1 = sP[r][1], p2 = sP[r][2], p3 = sP[r][3];
        const float* v0 = &sV[(node * 4 + 0) * 65 + c0];
        const float* v1 = &sV[(node * 4 + 1) * 65 + c0];
        const float* v2 = &sV[(node * 4 + 2) * 65 + c0];
        const float* v3 = &sV[(node * 4 + 3) * 65 + c0];
        float* orow = out + ((size_t)blockIdx.x * 64 + r) * Hc + c0;
        for (int c = 0; c < 32; ++c)
            orow[c] = p0 * v0[c] + p1 * v1[c] + p2 * v2[c] + p3 * v3[c];
    }
}

// ---------------------------------------------------------------------------
extern "C" void kernel_launch(void* const* d_in, const int* in_sizes, int n_in,
                              void* d_out, int out_size, void* d_ws, size_t ws_size,
                              hipStream_t stream) {
    const int*   cfg      = (const int*)  d_in[0];   // [B,E,2]
    const float* hidden   = (const float*)d_in[1];   // [B,N,IP,H]
    const float* node_fts = (const float*)d_in[2];   // [B,N,IP,H]
    const float* edge_fts = (const float*)d_in[3];   // [B,E,EF]
    const float* W_nh     = (const float*)d_in[4];   // [128,64]
    const float* b_nh     = (const float*)d_in[5];   // [64]
    const float* W_e      = (const float*)d_in[6];   // [64,1]
    const float* b_e      = (const float*)d_in[7];   // [1]
    const float* W_q      = (const float*)d_in[8];
    const float* b_q      = (const float*)d_in[9];
    const float* W_k      = (const float*)d_in[10];
    const float* b_k      = (const float*)d_in[11];
    const float* W_v      = (const float*)d_in[12];
    const float* b_v      = (const float*)d_in[13];
    float* out = (float*)d_out;

    float* nh_fused = (float*)d_ws;          // 10,240,000 floats (41 MB)
    float* agg      = nh_fused + NH_ELEMS;   // 10,240,000 floats (41 MB)

    // K1: 2500 blocks x 128 threads = 160000 rows
    k_nh_fused<<<ROWS / 64, 128, 0, stream>>>(node_fts, hidden, W_nh, b_nh, nh_fused);
    // K2: 10,240,000 floats = 2,560,000 float4 = 10000 blocks x 256 threads
    k_init_neg_inf<<<(int)(NH_ELEMS / 4 / 256), 256, 0, stream>>>((float4*)agg);
    // K3: B*E = 320000 edges, 8 waves/block -> 40000 blocks
    k_edge_scatter<<<(Bc * Ec) / 8, 256, 0, stream>>>(cfg, edge_fts, W_e, b_e,
                                                      nh_fused, agg);
    // K4: 2500 blocks x 128 threads
    k_qkv_attn<<<ROWS / 64, 128, 0, stream>>>(agg, W_q, b_q, W_k, b_k, W_v, b_v, out);
}